// Net_20151986553545
// MI455X (gfx1250) — compile-verified
//
#include <hip/hip_runtime.h>
#include <hip/hip_bf16.h>

typedef float v2f __attribute__((ext_vector_type(2)));
typedef float v8f __attribute__((ext_vector_type(8)));

#define BN 4
#define WSTRIDE 268   // max padded Cin (265 -> 268), row stride in LDS weight tile
#define NT 2          // out-channel tiles (16 each) per wave: A-operand reuse x2

// ---------------------------------------------------------------------------
// fused normalize: mean, radius, write normalized coords, save c/rad
// ---------------------------------------------------------------------------
__global__ void k_normalize(const float* __restrict__ xyz, float* __restrict__ xyzn,
                            float* __restrict__ cbuf, float* __restrict__ radbuf, int Np) {
  __shared__ float red[256];
  __shared__ float cm[3];
  __shared__ float srad;
  int b = blockIdx.x, tid = threadIdx.x;
  const float* xb = xyz + (long)b * 3 * Np;
  for (int d = 0; d < 3; d++) {
    float s = 0.f;
    for (int p = tid; p < Np; p += 256) s += xb[(long)d * Np + p];
    red[tid] = s; __syncthreads();
    for (int st = 128; st > 0; st >>= 1) { if (tid < st) red[tid] += red[tid + st]; __syncthreads(); }
    if (tid == 0) cm[d] = red[0] / (float)Np;
    __syncthreads();
  }
  float mx = 0.f;
  for (int p = tid; p < Np; p += 256) {
    float dx = xb[p] - cm[0], dy = xb[Np + p] - cm[1], dz = xb[2L * Np + p] - cm[2];
    mx = fmaxf(mx, sqrtf(dx * dx + dy * dy + dz * dz));
  }
  red[tid] = mx; __syncthreads();
  for (int st = 128; st > 0; st >>= 1) { if (tid < st) red[tid] = fmaxf(red[tid], red[tid + st]); __syncthreads(); }
  if (tid == 0) {
    srad = red[0]; radbuf[b] = red[0];
    cbuf[b * 3 + 0] = cm[0]; cbuf[b * 3 + 1] = cm[1]; cbuf[b * 3 + 2] = cm[2];
  }
  __syncthreads();
  float inv = 1.f / srad;
  float* yb = xyzn + (long)b * 3 * Np;
  for (int p = tid; p < Np; p += 256) {
    yb[p]           = (xb[p] - cm[0]) * inv;
    yb[Np + p]      = (xb[Np + p] - cm[1]) * inv;
    yb[2L * Np + p] = (xb[2L * Np + p] - cm[2]) * inv;
  }
}

// ---------------------------------------------------------------------------
// 1x1 conv as GEMM with V_WMMA_F32_16X16X4_F32, N-blocked x2 for A reuse.
// X: [B, Cin, Npos] (batch stride xb). W: [Cout, Cin] row-major, staged in LDS
// (zero-padded) so the K-loop is branch-free. Bias folded into acc init,
// ReLU fused into two float4 stores per lane per tile.
// xe_mode: virtual input = [repeat(X,2,points), code]; the code channel's
// contribution is code(p)*W[och][264] with code parity == v&1, folded at store.
// ---------------------------------------------------------------------------
__global__ void k_conv1x1_wmma(const float* __restrict__ X, long xb,
                               const float* __restrict__ W, const float* __restrict__ Bv,
                               float* __restrict__ Y, long yb,
                               int Cin, int Cout, int Npos, int relu, int xe_mode,
                               int np_src) {
  __shared__ float sW[16 * NT * WSTRIDE];
  __shared__ float sBias[16 * NT];
  int b = blockIdx.z;
  int tid = threadIdx.x;
  int outb = blockIdx.y * (16 * NT);
  int KP = (Cin + 3) & ~3;

  for (int t = tid; t < 16 * NT * KP; t += 256) {
    int r = t / KP, k = t - r * KP;
    int oc = outb + r;
    sW[r * WSTRIDE + k] = (oc < Cout && k < Cin) ? W[(long)oc * Cin + k] : 0.f;
  }
  if (tid < 16 * NT) {
    int oc = outb + tid;
    sBias[tid] = (oc < Cout) ? Bv[oc] : 0.f;
  }
  __syncthreads();

  int wave = tid >> 5, lane = tid & 31, half = lane >> 4, l16 = lane & 15;
  int ntiles = Npos >> 4;
  int ptile = blockIdx.x * 8 + wave;
  if (ptile >= ntiles) return;              // wave-uniform; no barriers after this
  int pbase = ptile << 4;
  int pA = pbase + l16;

  long cstride = xe_mode ? (long)np_src : (long)Npos;
  const float* xp = X + (long)b * xb + (xe_mode ? (pA >> 1) : pA) +
                    (long)(half << 1) * cstride;
  const float* w0 = sW + l16 * WSTRIDE + (half << 1);

  v8f acc[NT];
#pragma unroll
  for (int nt = 0; nt < NT; nt++) {
    float bias = sBias[nt * 16 + l16];
    acc[nt] = (v8f){bias, bias, bias, bias, bias, bias, bias, bias};
  }

  int kmain = xe_mode ? (Cin - 1) : (Cin & ~3);  // xe: 264 data channels
#pragma unroll 4
  for (int k0 = 0; k0 < kmain; k0 += 4) {
    v2f a;
    a.x = xp[0];
    a.y = xp[cstride];
#pragma unroll
    for (int nt = 0; nt < NT; nt++) {
      v2f bm;
      bm.x = w0[nt * 16 * WSTRIDE];
      bm.y = w0[nt * 16 * WSTRIDE + 1];
      acc[nt] = __builtin_amdgcn_wmma_f32_16x16x4_f32(false, a, false, bm, (short)0,
                                                      acc[nt], false, false);
    }
    xp += 4 * cstride;
    w0 += 4;
  }
  if (!xe_mode && (Cin & 3)) {                   // tail (layer0: Cin = 3)
    int ka = kmain + (half << 1);
    v2f a;
    a.x = (ka < Cin) ? xp[0] : 0.f;
    a.y = (ka + 1 < Cin) ? xp[cstride] : 0.f;
#pragma unroll
    for (int nt = 0; nt < NT; nt++) {
      v2f bm;
      bm.x = w0[nt * 16 * WSTRIDE];              // zero-padded in LDS
      bm.y = w0[nt * 16 * WSTRIDE + 1];
      acc[nt] = __builtin_amdgcn_wmma_f32_16x16x4_f32(false, a, false, bm, (short)0,
                                                      acc[nt], false, false);
    }
  }
  if (xe_mode) {                                 // fold code channel analytically
#pragma unroll
    for (int nt = 0; nt < NT; nt++) {
      float w264 = sW[(nt * 16 + l16) * WSTRIDE + (Cin - 1)];
#pragma unroll
      for (int v = 0; v < 8; v++) acc[nt][v] += ((v & 1) ? 0.2f : -0.2f) * w264;
    }
  }
#pragma unroll
  for (int nt = 0; nt < NT; nt++) {
    int och = outb + nt * 16 + l16;
    if (och < Cout) {
      float* yp = Y + (long)b * yb + (long)och * Npos + pbase + (half << 3);
      float4 lo, hi;
      if (relu) {
        lo = make_float4(fmaxf(acc[nt][0], 0.f), fmaxf(acc[nt][1], 0.f),
                         fmaxf(acc[nt][2], 0.f), fmaxf(acc[nt][3], 0.f));
        hi = make_float4(fmaxf(acc[nt][4], 0.f), fmaxf(acc[nt][5], 0.f),
                         fmaxf(acc[nt][6], 0.f), fmaxf(acc[nt][7], 0.f));
      } else {
        lo = make_float4(acc[nt][0], acc[nt][1], acc[nt][2], acc[nt][3]);
        hi = make_float4(acc[nt][4], acc[nt][5], acc[nt][6], acc[nt][7]);
      }
      *(float4*)(yp)     = lo;
      *(float4*)(yp + 4) = hi;
    }
  }
}

// ---------------------------------------------------------------------------
// KNN: per-thread top-K kept fully in VGPRs (static-index bubble insert),
// candidate tiles staged in LDS. C is a template param so qf[] unrolls.
// ---------------------------------------------------------------------------
template <int KK, int C>
__global__ void k_knn(const float* __restrict__ Q, long qb,
                      const float* __restrict__ P, long pb,
                      int Nq, int Nm, int* __restrict__ out) {
  const int TILE = 64;
  __shared__ float sp[C * TILE];
  int b = blockIdx.y;
  int q = blockIdx.x * blockDim.x + threadIdx.x;
  float qf[C];
  if (q < Nq) {
#pragma unroll
    for (int c = 0; c < C; c++) qf[c] = Q[(long)b * qb + (long)c * Nq + q];
  }
  float bd[KK]; int bi[KK];
#pragma unroll
  for (int i = 0; i < KK; i++) { bd[i] = 3.4e38f; bi[i] = 0; }

  for (int m0 = 0; m0 < Nm; m0 += TILE) {
    int cnt = min(TILE, Nm - m0);
    __syncthreads();
    for (int t = threadIdx.x; t < C * TILE; t += blockDim.x) {
      int c = t / TILE, m = t % TILE;
      sp[t] = (m < cnt) ? P[(long)b * pb + (long)c * Nm + m0 + m] : 0.f;
    }
    __syncthreads();
    if (q < Nq) {
      for (int m = 0; m < cnt; m++) {
        float d = 0.f;
#pragma unroll
        for (int c = 0; c < C; c++) { float t = qf[c] - sp[c * TILE + m]; d = fmaf(t, t, d); }
        if (d < bd[KK - 1]) {              // rare path: bubble into sorted list
          bd[KK - 1] = d; bi[KK - 1] = m0 + m;
#pragma unroll
          for (int i = KK - 1; i > 0; i--) {
            if (bd[i] < bd[i - 1]) {
              float td = bd[i]; bd[i] = bd[i - 1]; bd[i - 1] = td;
              int ti = bi[i]; bi[i] = bi[i - 1]; bi[i - 1] = ti;
            }
          }
        }
      }
    }
  }
  if (q < Nq) {
#pragma unroll
    for (int i = 0; i < KK; i++) out[((long)b * Nq + q) * KK + i] = bi[i];
  }
}

// ---------------------------------------------------------------------------
// Fused dense EdgeConv: gather 16 nbrs, 3 MLPs (48->12, 36->12, 48->12, ReLU)
// in registers with weights in LDS, max over 16 neighbors via shfl_xor
// butterflies (wave32: two 16-lane neighbor groups per wave).
// Output channels: [h3(12) | h2(12) | h1(12) | ctr(24)] = 60.
// ---------------------------------------------------------------------------
__global__ void k_edge_conv(const float* __restrict__ X, long xb,
                            const int* __restrict__ idx,
                            const float* __restrict__ W1, const float* __restrict__ B1,
                            const float* __restrict__ W2, const float* __restrict__ B2,
                            const float* __restrict__ W3, const float* __restrict__ B3,
                            float* __restrict__ out, long ob, int Np) {
  __shared__ float sW1[12 * 48], sW2[12 * 36], sW3[12 * 48], sB[36];
  __shared__ float sctr[8][24];
  int tid = threadIdx.x, b = blockIdx.y;
  for (int t = tid; t < 576; t += 128) sW1[t] = W1[t];
  for (int t = tid; t < 432; t += 128) sW2[t] = W2[t];
  for (int t = tid; t < 576; t += 128) sW3[t] = W3[t];
  if (tid < 12) sB[tid] = B1[tid];
  else if (tid < 24) sB[tid] = B2[tid - 12];
  else if (tid < 36) sB[tid] = B3[tid - 24];

  int pl = tid >> 4, j = tid & 15;
  int p0 = blockIdx.x * 8;
  for (int t = tid; t < 8 * 24; t += 128) {
    int pp = t / 24, c = t % 24, p = p0 + pp;
    sctr[pp][c] = (p < Np) ? X[(long)b * xb + (long)c * Np + p] : 0.f;
  }
  __syncthreads();

  int p = p0 + pl;
  int valid = (p < Np);
  int nidx = valid ? idx[((long)b * Np + p) * 16 + j] : 0;
  float ctr[24], nb[24], h1[12], h2[12], h3[12];
#pragma unroll
  for (int c = 0; c < 24; c++) {
    ctr[c] = sctr[pl][c];
    nb[c] = valid ? X[(long)b * xb + (long)c * Np + nidx] : 0.f;
  }
#pragma unroll
  for (int o = 0; o < 12; o++) {               // MLP1: [ctr, nbr-ctr] -> 12
    float s = sB[o];
    for (int c = 0; c < 24; c++) s = fmaf(sW1[o * 48 + c], ctr[c], s);
    for (int c = 0; c < 24; c++) s = fmaf(sW1[o * 48 + 24 + c], nb[c] - ctr[c], s);
    h1[o] = fmaxf(s, 0.f);
  }
#pragma unroll
  for (int o = 0; o < 12; o++) {               // MLP2: [h1, ctr] -> 12
    float s = sB[12 + o];
    for (int c = 0; c < 12; c++) s = fmaf(sW2[o * 36 + c], h1[c], s);
    for (int c = 0; c < 24; c++) s = fmaf(sW2[o * 36 + 12 + c], ctr[c], s);
    h2[o] = fmaxf(s, 0.f);
  }
#pragma unroll
  for (int o = 0; o < 12; o++) {               // MLP3: [h2, h1, ctr] -> 12
    float s = sB[24 + o];
    for (int c = 0; c < 12; c++) s = fmaf(sW3[o * 48 + c], h2[c], s);
    for (int c = 0; c < 12; c++) s = fmaf(sW3[o * 48 + 12 + c], h1[c], s);
    for (int c = 0; c < 24; c++) s = fmaf(sW3[o * 48 + 24 + c], ctr[c], s);
    h3[o] = fmaxf(s, 0.f);
  }
  for (int off = 8; off >= 1; off >>= 1) {     // max over 16 neighbors (in-wave)
#pragma unroll
    for (int i = 0; i < 12; i++) {
      h3[i] = fmaxf(h3[i], __shfl_xor(h3[i], off, 32));
      h2[i] = fmaxf(h2[i], __shfl_xor(h2[i], off, 32));
      h1[i] = fmaxf(h1[i], __shfl_xor(h1[i], off, 32));
    }
  }
  if (valid && j == 0) {
#pragma unroll
    for (int i = 0; i < 12; i++) {
      out[(long)b * ob + (long)(i)      * Np + p] = h3[i];
      out[(long)b * ob + (long)(12 + i) * Np + p] = h2[i];
      out[(long)b * ob + (long)(24 + i) * Np + p] = h1[i];
    }
#pragma unroll
    for (int c = 0; c < 24; c++)
      out[(long)b * ob + (long)(36 + c) * Np + p] = ctr[c];
  }
}

// ---------------------------------------------------------------------------
// Feature matching (level 1)
// ---------------------------------------------------------------------------
__global__ void k_zero(float* p, int n) {
  int i = blockIdx.x * blockDim.x + threadIdx.x;
  if (i < n) p[i] = 0.f;
}

__global__ void k_fm_stats(const float* __restrict__ xyz, const float* __restrict__ pxyz,
                           const float* __restrict__ feat, long fb,
                           const float* __restrict__ pfeat, long pfb,
                           const int* __restrict__ idx3,
                           float* __restrict__ dxyz, float* __restrict__ dfeat,
                           float* __restrict__ hacc, int Nq, int Nm) {
  __shared__ float r1[256], r2[256];
  int b = blockIdx.y, tid = threadIdx.x;
  int n = blockIdx.x * blockDim.x + tid;
  float mnx = 0.f, mnf = 0.f;
  if (n < Nq) {
    mnx = 3.4e38f; mnf = 3.4e38f;
    for (int k = 0; k < 3; k++) {
      int id = idx3[((long)b * Nq + n) * 3 + k];
      float dx = 0.f;
      for (int c = 0; c < 3; c++) {
        float t = xyz[(long)b * 3 * Nq + (long)c * Nq + n] -
                  pxyz[(long)b * 3 * Nm + (long)c * Nm + id];
        dx = fmaf(t, t, dx);
      }
      float df = 0.f;
      for (int c = 0; c < 264; c++) {
        float t = feat[(long)b * fb + (long)c * Nq + n] -
                  pfeat[(long)b * pfb + (long)c * Nm + id];
        df = fmaf(t, t, df);
      }
      dxyz[((long)b * Nq + n) * 3 + k] = dx;
      dfeat[((long)b * Nq + n) * 3 + k] = df;
      mnx = fminf(mnx, dx); mnf = fminf(mnf, df);
    }
  }
  r1[tid] = mnx; r2[tid] = mnf; __syncthreads();
  for (int st = 128; st > 0; st >>= 1) {
    if (tid < st) { r1[tid] += r1[tid + st]; r2[tid] += r2[tid + st]; }
    __syncthreads();
  }
  if (tid == 0) { atomicAdd(&hacc[b * 2 + 0], r1[0]); atomicAdd(&hacc[b * 2 + 1], r2[0]); }
}

__global__ void k_fm_apply(const float* __restrict__ dxyz, const float* __restrict__ dfeat,
                           const float* __restrict__ hacc, float* __restrict__ wn, int Nq) {
  int b = blockIdx.y;
  int n = blockIdx.x * blockDim.x + threadIdx.x;
  if (n >= Nq) return;
  float hx = hacc[b * 2 + 0] / (float)Nq * 0.5f;   // h/2
  float hf = hacc[b * 2 + 1] / (float)Nq * 0.5f;
  float w[3]; float s = 0.f;
  for (int k = 0; k < 3; k++) {
    float a = expf(-dxyz[((long)b * Nq + n) * 3 + k] / hx);
    float c = expf(-dfeat[((long)b * Nq + n) * 3 + k] / hf);
    w[k] = a * c;
    s += w[k] + 1e-5f;
  }
  for (int k = 0; k < 3; k++) wn[((long)b * Nq + n) * 3 + k] = w[k] / s;
}

__global__ void k_fm_add(float* __restrict__ feat, long fb,
                         const float* __restrict__ pfeat, long pfb,
                         const int* __restrict__ idx3, const float* __restrict__ wn,
                         int Nq, int Nm) {
  int b = blockIdx.y;
  long t = (long)blockIdx.x * blockDim.x + threadIdx.x;
  if (t >= (long)264 * Nq) return;
  int c = (int)(t / Nq), n = (int)(t % Nq);
  float s = 0.f;
  for (int k = 0; k < 3; k++) {
    int id = idx3[((long)b * Nq + n) * 3 + k];
    s += wn[((long)b * Nq + n) * 3 + k] * pfeat[(long)b * pfb + (long)c * Nm + id];
  }
  feat[(long)b * fb + (long)c * Nq + n] += 0.2f * s;
}

// ---------------------------------------------------------------------------
// residual + denormalize: dst = (o3 + repeat(xyzn)) * rad + c
// ---------------------------------------------------------------------------
__global__ void k_finalize(const float* __restrict__ o3, const float* __restrict__ xyzn,
                           const float* __restrict__ cbuf, const float* __restrict__ radbuf,
                           float* __restrict__ dst, int Np, int P2) {
  int b = blockIdx.y;
  int t = blockIdx.x * blockDim.x + threadIdx.x;
  if (t >= 3 * P2) return;
  int d = t / P2, j = t % P2;
  float v = o3[(long)b * 3 * P2 + (long)d * P2 + j] +
            xyzn[(long)b * 3 * Np + (long)d * Np + (j >> 1)];
  dst[(long)b * 3 * P2 + (long)d * P2 + j] = v * radbuf[b] + cbuf[b * 3 + d];
}

// ---------------------------------------------------------------------------
// host orchestration
// ---------------------------------------------------------------------------
extern "C" void kernel_launch(void* const* d_in, const int* in_sizes, int n_in,
                              void* d_out, int out_size, void* d_ws, size_t ws_size,
                              hipStream_t stream) {
  const float* xyz0 = (const float*)d_in[0];

  float* ws = (float*)d_ws;
  size_t off = 0;
  auto A = [&](size_t n) { float* p = ws + off; off += (n + 15) & ~(size_t)15; return p; };
  float* XYZN = A((size_t)BN * 3 * 4096);
  float* XYZ1 = A((size_t)BN * 3 * 4096);
  float* CBUF = A(BN * 3);
  float* RAD  = A(BN);
  float* FA   = A((size_t)BN * 264 * 2048);   // level-0 feats (persist for FM)
  float* FB   = A((size_t)BN * 264 * 4096);   // level-1 feats
  float* XP   = A((size_t)BN * 24 * 4096);    // prep outputs
  int*   IDX  = (int*)A((size_t)BN * 4096 * 16);
  int*   IDX3 = (int*)A((size_t)BN * 4096 * 3);
  float* DXY  = A((size_t)BN * 4096 * 3);
  float* DFT  = A((size_t)BN * 4096 * 3);
  float* WN   = A((size_t)BN * 4096 * 3);
  float* HACC = A(16);
  float* HBA  = A((size_t)BN * 128 * 8192);
  float* HBB  = A((size_t)BN * 128 * 8192);
  float* OUTP = A((size_t)BN * 3 * 8192);

  auto conv = [&](const float* X, long xb, const float* Wp, const float* Bp,
                  float* Y, long yb, int Cin, int Cout, int Npos, int relu,
                  int xe, int np_src) {
    int ntiles = Npos / 16;
    dim3 g((ntiles + 7) / 8, (Cout + 16 * NT - 1) / (16 * NT), BN);
    k_conv1x1_wmma<<<g, 256, 0, stream>>>(X, xb, Wp, Bp, Y, yb, Cin, Cout, Npos,
                                          relu, xe, np_src);
  };

  for (int l = 0; l < 2; l++) {
    int Np = 2048 << l;
    int P2 = Np * 2;
    const float* xyzc = (l == 0) ? xyz0 : XYZ1;
    float* FEAT = (l == 0) ? FA : FB;
    long fbs = (long)264 * Np;
    // JAX pytree (sorted-key) leaf order per level:
    // edge1..edge4 (W,b x3 each) = 0..23, fc1=24/25, fc2=26/27, layer0=28/29,
    // prep2=30/31, prep3=32/33, prep4=34/35, up1=36/37, up2=38/39
    int base = 1 + l * 40;
    auto Wp = [&](int j) { return (const float*)d_in[base + j]; };

    k_normalize<<<dim3(BN), 256, 0, stream>>>(xyzc, XYZN, CBUF, RAD, Np);

    // layer0: 3 -> 24, no activation; x0 lives at channels 240..263
    conv(XYZN, 3L * Np, Wp(28), Wp(29), FEAT + 240L * Np, fbs, 3, 24, Np, 0, 0, Np);

    dim3 gk((Np + 127) / 128, BN);
    dim3 ge(Np / 8, BN);

    // edge1 on x0 -> channels 180..239
    {
      const float* x0 = FEAT + 240L * Np;
      k_knn<16, 24><<<gk, 128, 0, stream>>>(x0, fbs, x0, fbs, Np, Np, IDX);
      k_edge_conv<<<ge, 128, 0, stream>>>(x0, fbs, IDX, Wp(0), Wp(1), Wp(2), Wp(3),
                                          Wp(4), Wp(5), FEAT + 180L * Np, fbs, Np);
    }
    // prep2 (84 ch window 180..263) + edge2 -> channels 120..179
    conv(FEAT + 180L * Np, fbs, Wp(30), Wp(31), XP, 24L * Np, 84, 24, Np, 1, 0, Np);
    k_knn<16, 24><<<gk, 128, 0, stream>>>(XP, 24L * Np, XP, 24L * Np, Np, Np, IDX);
    k_edge_conv<<<ge, 128, 0, stream>>>(XP, 24L * Np, IDX, Wp(6), Wp(7), Wp(8), Wp(9),
                                        Wp(10), Wp(11), FEAT + 120L * Np, fbs, Np);
    // prep3 (144 ch window 120..263) + edge3 -> channels 60..119
    conv(FEAT + 120L * Np, fbs, Wp(32), Wp(33), XP, 24L * Np, 144, 24, Np, 1, 0, Np);
    k_knn<16, 24><<<gk, 128, 0, stream>>>(XP, 24L * Np, XP, 24L * Np, Np, Np, IDX);
    k_edge_conv<<<ge, 128, 0, stream>>>(XP, 24L * Np, IDX, Wp(12), Wp(13), Wp(14), Wp(15),
                                        Wp(16), Wp(17), FEAT + 60L * Np, fbs, Np);
    // prep4 (204 ch window 60..263) + edge4 -> channels 0..59
    conv(FEAT + 60L * Np, fbs, Wp(34), Wp(35), XP, 24L * Np, 204, 24, Np, 1, 0, Np);
    k_knn<16, 24><<<gk, 128, 0, stream>>>(XP, 24L * Np, XP, 24L * Np, Np, Np, IDX);
    k_edge_conv<<<ge, 128, 0, stream>>>(XP, 24L * Np, IDX, Wp(18), Wp(19), Wp(20), Wp(21),
                                        Wp(22), Wp(23), FEAT, fbs, Np);

    // feature matching against previous level
    if (l == 1) {
      long pfb = 264L * 2048;
      k_knn<3, 3><<<gk, 128, 0, stream>>>(xyzc, 3L * Np, xyz0, 3L * 2048, Np, 2048, IDX3);
      k_zero<<<1, 64, 0, stream>>>(HACC, 16);
      dim3 gf((Np + 255) / 256, BN);
      k_fm_stats<<<gf, 256, 0, stream>>>(xyzc, xyz0, FEAT, fbs, FA, pfb, IDX3,
                                         DXY, DFT, HACC, Np, 2048);
      k_fm_apply<<<gf, 256, 0, stream>>>(DXY, DFT, HACC, WN, Np);
      dim3 ga(((long)264 * Np + 255) / 256, BN);
      k_fm_add<<<ga, 256, 0, stream>>>(FEAT, fbs, FA, pfb, IDX3, WN, Np, 2048);
    }

    // upsample MLP chain (up1 fuses repeat+code via xe_mode)
    conv(FEAT, fbs, Wp(36), Wp(37), HBA, 128L * P2, 265, 128, P2, 1, 1, Np);
    conv(HBA, 128L * P2, Wp(38), Wp(39), HBB, 128L * P2, 128, 128, P2, 1, 0, P2);
    conv(HBB, 128L * P2, Wp(24), Wp(25), HBA, 64L * P2, 128, 64, P2, 1, 0, P2);
    conv(HBA, 64L * P2, Wp(26), Wp(27), OUTP, 3L * P2, 64, 3, P2, 0, 0, P2);

    float* dst = (l == 0) ? XYZ1 : (float*)d_out;
    dim3 gz((3 * P2 + 255) / 256, BN);
    k_finalize<<<gz, 256, 0, stream>>>(OUTP, XYZN, CBUF, RAD, dst, Np, P2);
  }
  (void)in_sizes; (void)n_in; (void)out_size; (void)ws_size;
}